// RGCN_9749575762687
// MI455X (gfx1250) — compile-verified
//
#include <hip/hip_runtime.h>

// RGCN (2-layer, basis decomposition) for MI455X / gfx1250.
// Per-relation scatter into a 20MB L2-resident bucket, then f32 WMMA GEMM
// (V_WMMA_F32_16X16X4_F32). B operand staged TRANSPOSED in LDS so each
// fragment is one conflict-free ds_load_b64 straight into an even VGPR pair.

#define HDIM 256
#define RREL 16
#define KC   64              // k-chunk staged in LDS
#define LDST 68              // padded LDS row stride (words) for ldsBT[col][k]:
                             // bank = (4*col + k) % 64 -> conflict-free b64 fragment reads

typedef __attribute__((ext_vector_type(2))) float v2f;
typedef __attribute__((ext_vector_type(8))) float v8f;

// ---------------- utility kernels ----------------

__global__ __launch_bounds__(256) void zero_kernel(float* __restrict__ p, long n4) {
  // n4 = number of float4 elements
  float4* p4 = (float4*)p;
  long i = (long)blockIdx.x * blockDim.x + threadIdx.x;
  long stride = (long)gridDim.x * blockDim.x;
  float4 z = make_float4(0.f, 0.f, 0.f, 0.f);
  for (; i < n4; i += stride) p4[i] = z;
}

__global__ __launch_bounds__(256) void degree_kernel(const int* __restrict__ dst,
                                                     float* __restrict__ deg, int E) {
  int e = blockIdx.x * 256 + threadIdx.x;
  if (e < E) atomicAdd(&deg[dst[e]], 1.0f);
}

__global__ __launch_bounds__(256) void finalize_norm_kernel(float* __restrict__ norm, int N) {
  int n = blockIdx.x * 256 + threadIdx.x;
  if (n < N) {
    float d = norm[n];
    norm[n] = (d > 0.0f) ? (1.0f / d) : 0.0f;   // matches where(deg>0, 1/max(deg,1), 0)
  }
}

__global__ __launch_bounds__(256) void gather_kernel(const float* __restrict__ emb,
                                                     const int* __restrict__ nid,
                                                     float* __restrict__ h, int N) {
  long t = (long)blockIdx.x * 256 + threadIdx.x;
  int n = (int)(t >> 8);
  int j = (int)(t & (HDIM - 1));
  if (n < N) h[t] = emb[(long)nid[n] * HDIM + j];
}

// W[r,i,j] = sum_b wcomp[r,b] * bases[b,i,j]
__global__ __launch_bounds__(256) void compute_W_kernel(const float* __restrict__ wcomp,
                                                        const float* __restrict__ bases,
                                                        float* __restrict__ W) {
  int t = blockIdx.x * 256 + threadIdx.x;       // < R*H*H = 1M
  int r = t >> 16;                              // H*H = 65536
  int ij = t & 0xFFFF;
  float acc = 0.0f;
#pragma unroll
  for (int b = 0; b < RREL; ++b)
    acc += wcomp[r * RREL + b] * bases[(long)b * HDIM * HDIM + ij];
  W[t] = acc;
}

// For edges with etype==rel: tmp[dst,:] += norm[dst] * X[src,:]
// One wave (32 lanes) per edge; 8 coalesced f32 atomics per lane.
__global__ __launch_bounds__(256) void scatter_kernel(const float* __restrict__ X,
                                                      const int* __restrict__ src,
                                                      const int* __restrict__ dst,
                                                      const int* __restrict__ etype,
                                                      const float* __restrict__ norm,
                                                      float* __restrict__ tmp,
                                                      int rel, int E) {
  int e = blockIdx.x * 8 + (threadIdx.x >> 5);
  int lane = threadIdx.x & 31;
  if (e >= E) return;
  if (etype[e] != rel) return;
  int s = src[e];
  int d = dst[e];
  float nm = norm[d];
  const float* xs = X + (long)s * HDIM;
  float* td = tmp + (long)d * HDIM;
#pragma unroll
  for (int j = lane; j < HDIM; j += 32)
    atomicAdd(&td[j], nm * xs[j]);
}

// ---------------- WMMA GEMM with transposed LDS-staged B ----------------
// Y[m, n0:n0+128] (+)= A[m, :] @ B[:, n0:n0+128].  A: [N x 256] row-major,
// B: [256 x 256] row-major.  8 waves/block: wave w owns M strip (blockIdx.x*8+w),
// all waves share the B k-chunk staged (transposed) in LDS.
// acc==0: C initialized from bias[col];  acc==1: C initialized from Y.
//
// Fragment layouts per CDNA5 ISA 7.12.2 (wave32):
//   A 16x4 f32:  lane l holds row m0+(l&15), k = kbase + 2*(l>>4) + {reg0,reg1}
//   B 4x16  f32: lane l holds col n+(l&15),  k = kbase + 2*(l>>4) + {reg0,reg1}
//   C 16x16 f32: lane l, reg v -> row m0 + v + 8*(l>>4), col n + (l&15)
__global__ __launch_bounds__(256)
void gemm16_kernel(const float* __restrict__ A, const float* __restrict__ B,
                   float* __restrict__ Y, const float* __restrict__ bias,
                   int acc, int nstrips) {
  __shared__ float ldsBT[128 * LDST];          // ldsBT[col][k], col in [0,128), k in [0,KC)

  const int tid  = threadIdx.x;
  const int lane = tid & 31;
  const int wave = tid >> 5;
  const int half = lane >> 4;
  const int lm   = lane & 15;
  const int strip = blockIdx.x * 8 + wave;     // wave-uniform
  const bool active = (strip < nstrips);       // wave-uniform -> EXEC stays all-1s
  const int m0 = strip << 4;                   // 16-row strip
  const int n0 = blockIdx.y << 7;              // 128-col half

  v8f c[8];
  if (active) {
    if (acc) {
#pragma unroll
      for (int t = 0; t < 8; ++t)
#pragma unroll
        for (int v = 0; v < 8; ++v)
          c[t][v] = Y[(long)(m0 + v + 8 * half) * HDIM + n0 + t * 16 + lm];
    } else {
#pragma unroll
      for (int t = 0; t < 8; ++t) {
        float bv = bias[n0 + t * 16 + lm];
#pragma unroll
        for (int v = 0; v < 8; ++v) c[t][v] = bv;
      }
    }
  }

  const float* Arow = A + (long)(m0 + lm) * HDIM;

  for (int kc = 0; kc < HDIM; kc += KC) {
    __syncthreads();
    // cooperative stage of B[kc:kc+KC, n0:n0+128] into LDS, transposed.
    // Global reads fully coalesced (float4 along columns); LDS stores scatter.
#pragma unroll
    for (int it = 0; it < 8; ++it) {
      int li   = it * 256 + tid;       // float4 index in [0, 2048)
      int row  = li >> 5;              // k row, [0, KC)
      int col4 = li & 31;              // col = col4*4, [0, 128)
      float4 vv = *(const float4*)(B + (long)(kc + row) * HDIM + n0 + col4 * 4);
      ldsBT[(col4 * 4 + 0) * LDST + row] = vv.x;
      ldsBT[(col4 * 4 + 1) * LDST + row] = vv.y;
      ldsBT[(col4 * 4 + 2) * LDST + row] = vv.z;
      ldsBT[(col4 * 4 + 3) * LDST + row] = vv.w;
    }
    __syncthreads();

    if (active) {
#pragma unroll
      for (int kk = 0; kk < KC; kk += 4) {
        const int ka = kk + 2 * half;          // even -> 8B-aligned LDS b64 load
        v2f a;
        a.x = Arow[kc + ka];
        a.y = Arow[kc + ka + 1];
#pragma unroll
        for (int t = 0; t < 8; ++t) {
          const int coll = t * 16 + lm;
          v2f b = *(const v2f*)&ldsBT[coll * LDST + ka];   // one ds_load_b64
          // 8 args: (neg_a, A, neg_b, B, c_mod, C, reuse_a, reuse_b)
          c[t] = __builtin_amdgcn_wmma_f32_16x16x4_f32(
              false, a, false, b, (short)0, c[t], false, false);
        }
      }
    }
  }

  if (active) {
#pragma unroll
    for (int t = 0; t < 8; ++t)
#pragma unroll
      for (int v = 0; v < 8; ++v)
        Y[(long)(m0 + v + 8 * half) * HDIM + n0 + t * 16 + lm] = c[t][v];
  }
}

// ---------------- launch ----------------

extern "C" void kernel_launch(void* const* d_in, const int* in_sizes, int n_in,
                              void* d_out, int out_size, void* d_ws, size_t ws_size,
                              hipStream_t stream) {
  (void)n_in; (void)out_size; (void)ws_size;

  const float* emb    = (const float*)d_in[0];
  const float* bases0 = (const float*)d_in[1];
  const float* wcomp0 = (const float*)d_in[2];
  const float* loopw0 = (const float*)d_in[3];
  const float* bias0  = (const float*)d_in[4];
  const float* bases1 = (const float*)d_in[5];
  const float* wcomp1 = (const float*)d_in[6];
  const float* loopw1 = (const float*)d_in[7];
  const float* bias1  = (const float*)d_in[8];
  const int* node_id  = (const int*)d_in[9];
  const int* esrc     = (const int*)d_in[10];
  const int* edst     = (const int*)d_in[11];
  const int* etype    = (const int*)d_in[12];

  const int  N  = in_sizes[9];   // 20000 (divisible by 16)
  const int  E  = in_sizes[10];  // 320000
  const long NH = (long)N * HDIM;
  const int  nstrips = N / 16;   // 1250

  // workspace layout (~66 MB total)
  char* ws = (char*)d_ws;
  float* h0   = (float*)ws;  ws += NH * sizeof(float);
  float* h1   = (float*)ws;  ws += NH * sizeof(float);
  float* tmp  = (float*)ws;  ws += NH * sizeof(float);
  float* Wall = (float*)ws;  ws += (long)RREL * HDIM * HDIM * sizeof(float);
  float* norm = (float*)ws;  ws += (long)((N + 3) & ~3) * sizeof(float);

  // edge norm (shared by both layers)
  zero_kernel<<<64, 256, 0, stream>>>(norm, (long)((N + 3) / 4));
  degree_kernel<<<(E + 255) / 256, 256, 0, stream>>>(edst, norm, E);
  finalize_norm_kernel<<<(N + 255) / 256, 256, 0, stream>>>(norm, N);

  // h = emb[node_id]
  gather_kernel<<<(int)((NH + 255) / 256), 256, 0, stream>>>(emb, node_id, h0, N);

  const float* basesL[2] = {bases0, bases1};
  const float* wcompL[2] = {wcomp0, wcomp1};
  const float* loopwL[2] = {loopw0, loopw1};
  const float* biasL[2]  = {bias0, bias1};
  float* Xb[2] = {h0, h1};
  float* Yb[2] = {h1, (float*)d_out};

  const dim3 ggrid((nstrips + 7) / 8, HDIM / 128);
  for (int l = 0; l < 2; ++l) {
    // W[r] = sum_b wcomp[r,b]*bases[b]
    compute_W_kernel<<<RREL * HDIM * HDIM / 256, 256, 0, stream>>>(wcompL[l], basesL[l], Wall);
    // Y = bias + X @ loop_w   (WMMA, init-from-bias mode)
    gemm16_kernel<<<ggrid, 256, 0, stream>>>(Xb[l], loopwL[l], Yb[l], biasL[l], 0, nstrips);
    // Y += sum_r bucket_r @ W[r]
    for (int r = 0; r < RREL; ++r) {
      zero_kernel<<<2048, 256, 0, stream>>>(tmp, NH / 4);
      scatter_kernel<<<(E + 7) / 8, 256, 0, stream>>>(Xb[l], esrc, edst, etype, norm, tmp, r, E);
      gemm16_kernel<<<ggrid, 256, 0, stream>>>(tmp, Wall + (long)r * HDIM * HDIM, Yb[l], nullptr, 1, nstrips);
    }
  }
}